// WaveletSpectralBlock_46162308498103
// MI455X (gfx1250) — compile-verified
//
#include <hip/hip_runtime.h>
#include <hip/hip_bf16.h>

typedef __attribute__((ext_vector_type(2))) float v2f;
typedef __attribute__((ext_vector_type(8))) float v8f;

namespace {
constexpr int kB  = 8;
constexpr int kC  = 128;
constexpr int kL  = 8192;
constexpr int kCH = 128;
constexpr int kS  = 16;
constexpr int kPad = kL / 2;          // 4096
constexpr int kLP  = kL + 2 * kPad;   // 16384 padded signal length
constexpr float kOmega0 = 6.0f;
constexpr float kBnEps  = 1e-5f;
}

// ---------------------------------------------------------------------------
// Stage 0a: generate wavelet bank wav[s][j] = exp(-0.5 ts^2) * cos(6 ts)
// ---------------------------------------------------------------------------
__global__ __launch_bounds__(256) void wsb_wavelet(const float* __restrict__ scales,
                                                   float* __restrict__ wav) {
  int idx = blockIdx.x * 256 + threadIdx.x;            // [0, 16*8192)
  int s = idx / kL;
  int j = idx - s * kL;
  float t  = -4.0f + (8.0f / 8191.0f) * (float)j;      // linspace(-4, 4, 8192)
  float ts = t / scales[s];
  wav[idx] = expf(-0.5f * ts * ts) * cosf(kOmega0 * ts);
}

// ---------------------------------------------------------------------------
// Stage 0b: channel mean with zero padding: xm_pad[b][p], p in [0,16384)
// ---------------------------------------------------------------------------
__global__ __launch_bounds__(256) void wsb_chanmean(const float* __restrict__ x,
                                                    float* __restrict__ xmp) {
  int idx = blockIdx.x * 256 + threadIdx.x;            // [0, 8*16384)
  int b = idx / kLP;
  int p = idx - b * kLP;
  float v = 0.0f;
  if (p >= kPad && p < kPad + kL) {
    int l = p - kPad;
    const float* xb = x + (size_t)b * kC * kL + l;
    float acc = 0.0f;
#pragma unroll 8
    for (int c = 0; c < kC; ++c) acc += xb[(size_t)c * kL];
    v = acc * (1.0f / (float)kC);
  }
  xmp[idx] = v;
}

// ---------------------------------------------------------------------------
// Stage 1: CWT as Toeplitz matmul with V_WMMA_F32_16X16X4_F32.
//   cwt[b][s][i] = sum_j wav[s][j] * xm_pad[b][i + j]
// One wave computes a 16(scales) x 64(positions) strip: 4 accumulators,
// K swept 0..8191 in steps of 4. A tile shared across the 4 WMMAs.
// Block = 128 threads (4 waves) -> 256 columns; grid = 8 batches * 32 blocks.
// ---------------------------------------------------------------------------
__global__ __launch_bounds__(128) void wsb_cwt_wmma(const float* __restrict__ wav,
                                                    const float* __restrict__ xmp,
                                                    float* __restrict__ cwt) {
  const int lane = threadIdx.x & 31;
  const int wave = threadIdx.x >> 5;
  const int b    = blockIdx.x >> 5;                    // 32 blocks per batch
  const int blk  = blockIdx.x & 31;
  const int i0   = blk * 256 + wave * 64;              // first output column
  const int row  = lane & 15;                          // M row / N column (lane%16)
  const int koff = (lane >> 4) * 2;                    // K sub-offset per half-wave

  const float* __restrict__ xb = xmp + (size_t)b * kLP;
  const float* __restrict__ wr = wav + (size_t)row * kL;

  v8f acc0 = {}, acc1 = {}, acc2 = {}, acc3 = {};

#pragma unroll 4
  for (int k = 0; k < kL; k += 4) {
    // A tile 16x4: lane<16 holds {K=k,k+1}, lane>=16 holds {K=k+2,k+3} (8B aligned)
    v2f a = *(const v2f*)(wr + k + koff);
    // B tiles 4x16 (Toeplitz): column n = lane%16, same K split as A
    const float* bp = xb + i0 + row + k + koff;
    v2f b0; b0.x = bp[0];  b0.y = bp[1];
    v2f b1; b1.x = bp[16]; b1.y = bp[17];
    v2f b2; b2.x = bp[32]; b2.y = bp[33];
    v2f b3; b3.x = bp[48]; b3.y = bp[49];
    acc0 = __builtin_amdgcn_wmma_f32_16x16x4_f32(false, a, false, b0, (short)0, acc0, false, false);
    acc1 = __builtin_amdgcn_wmma_f32_16x16x4_f32(false, a, false, b1, (short)0, acc1, false, false);
    acc2 = __builtin_amdgcn_wmma_f32_16x16x4_f32(false, a, false, b2, (short)0, acc2, false, false);
    acc3 = __builtin_amdgcn_wmma_f32_16x16x4_f32(false, a, false, b3, (short)0, acc3, false, false);
  }

  // D layout: VGPR r holds rows s=r (lanes 0-15) and s=r+8 (lanes 16-31), N=lane%16
  const int srow = (lane >> 4) * 8;
  const int col  = lane & 15;
  float* __restrict__ cb = cwt + (size_t)b * kS * kL;
#pragma unroll
  for (int r = 0; r < 8; ++r) {
    int s = srow + r;
    cb[(size_t)s * kL + i0 +  0 + col] = acc0[r];
    cb[(size_t)s * kL + i0 + 16 + col] = acc1[r];
    cb[(size_t)s * kL + i0 + 32 + col] = acc2[r];
    cb[(size_t)s * kL + i0 + 48 + col] = acc3[r];
  }
}

// ---------------------------------------------------------------------------
// Stage 2: 3x3 conv (1 -> 128 ch) + bias + BN + exact GELU + mean over scales.
// Block handles one (b, 128-wide l tile); cwt tile staged in LDS and reused
// by all 128 output channels. Weight/BN loads are wave-uniform -> scalar.
// ---------------------------------------------------------------------------
__global__ __launch_bounds__(256) void wsb_conv_gelu_mean(
    const float* __restrict__ cwt, const float* __restrict__ w,
    const float* __restrict__ bias, const float* __restrict__ gamma,
    const float* __restrict__ beta, const float* __restrict__ mean,
    const float* __restrict__ var, float* __restrict__ out) {
  __shared__ float tile[kS + 2][132];                  // rows s=-1..16, cols l0-1..l0+128

  const int b   = blockIdx.x >> 6;                     // 64 blocks per batch
  const int l0  = (blockIdx.x & 63) * 128;
  const int tid = threadIdx.x;

  for (int idx = tid; idx < (kS + 2) * 130; idx += 256) {
    int r = idx / 130;
    int c = idx - r * 130;
    int s = r - 1;
    int l = l0 + c - 1;
    float v = 0.0f;
    if (s >= 0 && s < kS && l >= 0 && l < kL)
      v = cwt[((size_t)b * kS + s) * kL + l];
    tile[r][c] = v;
  }
  __syncthreads();

  const int lpos = tid & 127;                          // output column in tile
  const int half = tid >> 7;                           // uniform within each wave

  for (int it = 0; it < kCH / 2; ++it) {
    const int ch = it * 2 + half;
    const float* wc = w + ch * 9;
    const float w00 = wc[0], w01 = wc[1], w02 = wc[2];
    const float w10 = wc[3], w11 = wc[4], w12 = wc[5];
    const float w20 = wc[6], w21 = wc[7], w22 = wc[8];
    const float bi  = bias[ch];
    const float inv = gamma[ch] * rsqrtf(var[ch] + kBnEps);
    const float sh  = beta[ch] - mean[ch] * inv;

    float acc = 0.0f;
#pragma unroll 4
    for (int s = 0; s < kS; ++s) {
      const float* r0 = &tile[s + 0][lpos];
      const float* r1 = &tile[s + 1][lpos];
      const float* r2 = &tile[s + 2][lpos];
      float v = w00 * r0[0] + w01 * r0[1] + w02 * r0[2]
              + w10 * r1[0] + w11 * r1[1] + w12 * r1[2]
              + w20 * r2[0] + w21 * r2[1] + w22 * r2[2];
      v = (v + bi) * inv + sh;                         // BN affine
      acc += 0.5f * v * (1.0f + erff(v * 0.70710678118654752f));  // exact GELU
    }
    out[((size_t)b * kCH + ch) * kL + l0 + lpos] = acc * (1.0f / (float)kS);
  }
}

// ---------------------------------------------------------------------------
// Launch
// ---------------------------------------------------------------------------
extern "C" void kernel_launch(void* const* d_in, const int* in_sizes, int n_in,
                              void* d_out, int out_size, void* d_ws, size_t ws_size,
                              hipStream_t stream) {
  const float* x      = (const float*)d_in[0];
  const float* scales = (const float*)d_in[1];
  const float* conv_w = (const float*)d_in[2];
  const float* conv_b = (const float*)d_in[3];
  const float* gamma  = (const float*)d_in[4];
  const float* beta   = (const float*)d_in[5];
  const float* mean   = (const float*)d_in[6];
  const float* var    = (const float*)d_in[7];
  float* out = (float*)d_out;

  // Workspace layout (floats): wav[16*8192] | xm_pad[8*16384] | cwt[8*16*8192]
  float* wav = (float*)d_ws;
  float* xmp = wav + (size_t)kS * kL;
  float* cwt = xmp + (size_t)kB * kLP;

  wsb_wavelet<<<(kS * kL) / 256, 256, 0, stream>>>(scales, wav);
  wsb_chanmean<<<(kB * kLP) / 256, 256, 0, stream>>>(x, xmp);
  wsb_cwt_wmma<<<kB * 32, 128, 0, stream>>>(wav, xmp, cwt);
  wsb_conv_gelu_mean<<<kB * (kL / 128), 256, 0, stream>>>(
      cwt, conv_w, conv_b, gamma, beta, mean, var, out);
}